// GraphAttentionLayer_20263655703137
// MI455X (gfx1250) — compile-verified
//
#include <hip/hip_runtime.h>

// ---------------------------------------------------------------------------
// GATv2 x2 for MI455X (gfx1250).
//   - fp32 WMMA (v_wmma_f32_16x16x4_f32) for the four dense transforms
//   - CSR build (int atomics + LDS scan) once per call
//   - dst-centric fused online-softmax aggregation: 1 wave per (dst, head),
//     1 lane per channel (wave32 == C=32), zero float atomics, uniform-branch
//     rescale skip (saves one v_exp per edge on ~90% of edges)
// ---------------------------------------------------------------------------

#define N_NODES 20000
#define N_EDGES 640000
#define EE_TOT  (N_EDGES + N_NODES)   // + self loops
#define HEADS   8
#define CH      32
#define FOUT    256                   // HEADS*CH
#define SLOPE   0.2f

typedef __attribute__((ext_vector_type(2))) float v2f;
typedef __attribute__((ext_vector_type(8))) float v8f;

// ---------------------------------------------------------------------------
// fp32 WMMA GEMM:  Y = X[M x K] * W[K x 256]   (M = 20000, 32-row tiles)
// Block = 256 threads = 8 waves; wave computes a 32x64 tile:
//   waves 0..3 -> Yl col-blocks 0..3 (weights Wl)
//   waves 4..7 -> Yr col-blocks 0..3 (weights Wr)
// Two A-fragments (rows m0..m0+15, m0+16..m0+31) share each B-fragment.
// A-frag (16x4 f32): lanes 0-15 carry K = k,k+1 ; lanes 16-31 K = k+2,k+3
// B-frag (4x16 f32): v0 = rows k / k+2, v1 = rows k+1 / k+3, col = lane&15
// C/D (16x16 f32):  vgpr j = row j (lanes 0-15) / row j+8 (lanes 16-31)
// ---------------------------------------------------------------------------
template <int K>
__global__ __launch_bounds__(256)
void wmma_gemm_kernel(const float* __restrict__ X,
                      const float* __restrict__ Wl,
                      const float* __restrict__ Wr,
                      float* __restrict__ Yl,
                      float* __restrict__ Yr) {
    const int wave    = threadIdx.x >> 5;     // 0..7
    const int lane    = threadIdx.x & 31;
    const int lane_lo = lane & 15;
    const int hi      = lane >> 4;            // 0 or 1
    const int m0      = blockIdx.x * 32;      // row tile base
    const int nb      = (wave & 3) * 64;      // col block base

    const float* __restrict__ W = (wave < 4) ? Wl : Wr;
    float* __restrict__       Y = (wave < 4) ? Yl : Yr;

    v8f acc[2][4] = {};

    const float* __restrict__ x0 = X + (m0 + lane_lo) * K + 2 * hi;
    const float* __restrict__ x1 = x0 + 16 * K;

    #pragma unroll 4
    for (int k = 0; k < K; k += 4) {
        v2f a0; a0.x = x0[k]; a0.y = x0[k + 1];
        v2f a1; a1.x = x1[k]; a1.y = x1[k + 1];

        const float* __restrict__ w0 = W + (k + 2 * hi) * FOUT + nb + lane_lo;
        const float* __restrict__ w1 = w0 + FOUT;   // row k+1(+2*hi)

        #pragma unroll
        for (int t = 0; t < 4; ++t) {
            v2f b; b.x = w0[t * 16]; b.y = w1[t * 16];
            acc[0][t] = __builtin_amdgcn_wmma_f32_16x16x4_f32(
                false, a0, false, b, (short)0, acc[0][t], false, false);
            acc[1][t] = __builtin_amdgcn_wmma_f32_16x16x4_f32(
                false, a1, false, b, (short)0, acc[1][t], false, false);
        }
    }

    #pragma unroll
    for (int r = 0; r < 2; ++r) {
        float* __restrict__ yrow = Y + (m0 + r * 16 + 8 * hi) * FOUT + nb + lane_lo;
        #pragma unroll
        for (int j = 0; j < 8; ++j) {
            #pragma unroll
            for (int t = 0; t < 4; ++t) {
                yrow[j * FOUT + t * 16] = acc[r][t][j];
            }
        }
    }
}

// ---------------------------------------------------------------------------
// CSR build helpers
// ---------------------------------------------------------------------------
__device__ __forceinline__ void edge_decode(int e, const long long* __restrict__ ei,
                                            int& s, int& d) {
    if (e < N_EDGES) {
        s = (int)ei[e];
        d = (int)ei[N_EDGES + e];
    } else {
        s = d = e - N_EDGES;   // self loop
    }
}

__global__ void izero_kernel(int* __restrict__ p, int n) {
    int i = blockIdx.x * blockDim.x + threadIdx.x;
    if (i < n) p[i] = 0;
}

__global__ __launch_bounds__(256)
void degree_kernel(const long long* __restrict__ ei, int* __restrict__ deg) {
    int e = blockIdx.x * blockDim.x + threadIdx.x;
    if (e >= EE_TOT) return;
    int s, d;
    edge_decode(e, ei, s, d);
    atomicAdd(&deg[d], 1);
}

// single-workgroup inclusive scan over N_NODES degrees -> exclusive rowptr
__global__ __launch_bounds__(1024)
void scan_kernel(const int* __restrict__ deg, int* __restrict__ rowptr) {
    __shared__ int buf[1024];
    __shared__ int carry_s;
    const int t = threadIdx.x;
    if (t == 0) { carry_s = 0; rowptr[0] = 0; }
    __syncthreads();
    for (int base = 0; base < N_NODES; base += 1024) {
        int i = base + t;
        buf[t] = (i < N_NODES) ? deg[i] : 0;
        __syncthreads();
        #pragma unroll
        for (int off = 1; off < 1024; off <<= 1) {
            int u = (t >= off) ? buf[t - off] : 0;
            __syncthreads();
            buf[t] += u;
            __syncthreads();
        }
        int carry = carry_s;
        if (i < N_NODES) rowptr[i + 1] = buf[t] + carry;
        __syncthreads();
        if (t == 1023) carry_s = carry + buf[1023];
        __syncthreads();
    }
}

__global__ __launch_bounds__(256)
void scatter_kernel(const long long* __restrict__ ei,
                    const int* __restrict__ rowptr,
                    int* __restrict__ cursor,
                    int* __restrict__ adj) {
    int e = blockIdx.x * blockDim.x + threadIdx.x;
    if (e >= EE_TOT) return;
    int s, d;
    edge_decode(e, ei, s, d);
    int pos = atomicAdd(&cursor[d], 1);
    adj[rowptr[d] + pos] = s;
}

// ---------------------------------------------------------------------------
// Fused GATv2 attention + aggregation (online softmax, dst-centric).
// grid = N_NODES blocks of 256 threads; wave w handles head w, lane = channel.
// Per incoming edge: one coalesced 128B gather of xl[src,h,:], a 5-step
// shfl_xor butterfly for the logit, and an online-rescaled accumulator.
// The logit / running-max comparison is wave-uniform, so the no-new-max
// fast path (one exp, no rescale) is a uniform branch.
// out[n,h,c] = sum_e alpha_e * xl[src_e,h,c] + bias[h,c]   (+ optional ELU)
// ---------------------------------------------------------------------------
__global__ __launch_bounds__(256)
void gat_aggregate_kernel(const int* __restrict__ rowptr,
                          const int* __restrict__ adj,
                          const float* __restrict__ xl,
                          const float* __restrict__ xr,
                          const float* __restrict__ att,   // [HEADS*CH]
                          const float* __restrict__ bias,  // [FOUT]
                          float* __restrict__ out,
                          int apply_elu) {
    const int n  = blockIdx.x;
    const int hc = threadIdx.x;       // h*32 + c : wave = head, lane = channel

    const float r = xr[n * FOUT + hc];
    const float w = att[hc];

    float m   = -__builtin_inff();
    float s   = 0.0f;
    float acc = 0.0f;

    const int beg = rowptr[n];
    const int end = rowptr[n + 1];
    for (int i = beg; i < end; ++i) {
        const int src = adj[i];
        const float xv = xl[src * FOUT + hc];
        float t = xv + r;
        t = t > 0.0f ? t : SLOPE * t;
        t *= w;
        // wave-wide sum -> logit l in all lanes
        #pragma unroll
        for (int off = 16; off > 0; off >>= 1) t += __shfl_xor(t, off, 32);
        const float l = t;
        if (l <= m) {                       // uniform fast path: max unchanged
            const float p = __expf(l - m);
            s   += p;
            acc += p * xv;
        } else {                            // new max: rescale (p == 1)
            const float scale = __expf(m - l);  // first iter: exp(-inf) = 0
            s   = s * scale + 1.0f;
            acc = acc * scale + xv;
            m   = l;
        }
    }

    float v = acc / s + bias[hc];
    if (apply_elu) v = v > 0.0f ? v : (__expf(v) - 1.0f);
    out[n * FOUT + hc] = v;
}

// ---------------------------------------------------------------------------
// launch
// ---------------------------------------------------------------------------
extern "C" void kernel_launch(void* const* d_in, const int* in_sizes, int n_in,
                              void* d_out, int out_size, void* d_ws, size_t ws_size,
                              hipStream_t stream) {
    const float*     X   = (const float*)d_in[0];        // [1,N,4,32] -> [N,128]
    const long long* ei  = (const long long*)d_in[1];    // [2,E] int64
    const float*     W1l = (const float*)d_in[2];
    const float*     W1r = (const float*)d_in[3];
    const float*     a1  = (const float*)d_in[4];        // [8,32]
    const float*     b1  = (const float*)d_in[5];        // [256]
    const float*     W2l = (const float*)d_in[6];
    const float*     W2r = (const float*)d_in[7];
    const float*     a2  = (const float*)d_in[8];
    const float*     b2  = (const float*)d_in[9];
    float*           out = (float*)d_out;                // [1,N,256]

    // workspace carve-up
    const int NF = N_NODES * FOUT;            // 5,120,000 floats
    float* xl = (float*)d_ws;
    float* xr = xl + NF;
    float* h1 = xr + NF;
    int*   deg    = (int*)(h1 + NF);
    int*   cursor = deg + N_NODES;
    int*   rowptr = cursor + N_NODES;         // N_NODES+1
    int*   adj    = rowptr + (N_NODES + 1);   // EE_TOT

    const int TB = 256;
    dim3 gemm_grid(N_NODES / 32);             // 625 row tiles (32 rows/tile)
    int  eg = (EE_TOT + TB - 1) / TB;         // per-edge grid
    int  ng = (N_NODES + TB - 1) / TB;

    // ---------------- CSR build (shared by both layers) ----------------
    izero_kernel  <<<ng, TB, 0, stream>>>(deg, N_NODES);
    izero_kernel  <<<ng, TB, 0, stream>>>(cursor, N_NODES);
    degree_kernel <<<eg, TB, 0, stream>>>(ei, deg);
    scan_kernel   <<<1, 1024, 0, stream>>>(deg, rowptr);
    scatter_kernel<<<eg, TB, 0, stream>>>(ei, rowptr, cursor, adj);

    // ---------------- layer 1 ----------------
    wmma_gemm_kernel<128><<<gemm_grid, TB, 0, stream>>>(X, W1l, W1r, xl, xr);
    gat_aggregate_kernel<<<N_NODES, TB, 0, stream>>>(rowptr, adj, xl, xr,
                                                     a1, b1, h1, /*elu=*/0);

    // ---------------- layer 2 ----------------
    wmma_gemm_kernel<256><<<gemm_grid, TB, 0, stream>>>(h1, W2l, W2r, xl, xr);
    gat_aggregate_kernel<<<N_NODES, TB, 0, stream>>>(rowptr, adj, xl, xr,
                                                     a2, b2, out, /*elu=*/1);
}